// EncoderInput_49349174231049
// MI455X (gfx1250) — compile-verified
//
#include <hip/hip_runtime.h>
#include <math.h>

typedef __attribute__((ext_vector_type(2))) float v2f;
typedef __attribute__((ext_vector_type(8))) float v8f;

#define T_DIM 64
#define N_DIM 81      // n
#define B_DIM 128
#define G_DIM 512     // 4*m
#define BT    8192    // B*T
#define ROWE  5184    // T*n

__device__ __forceinline__ float sigm_(float x) { return 1.0f / (1.0f + __expf(-x)); }

// ---- kernel 0: biasrow[b,j] = b_lstm[j] + sum_k h0[b,k]*U_lstm[k,j] ----------
__global__ __launch_bounds__(256)
void bias_gemm_kernel(const float* __restrict__ h0, const float* __restrict__ U_lstm,
                      const float* __restrict__ b_lstm, float* __restrict__ biasrow) {
    int idx = blockIdx.x * blockDim.x + threadIdx.x;   // 0..65535
    int b = idx >> 9, j = idx & 511;
    float acc = b_lstm[j];
    const float* hp = h0 + b * 128;
    #pragma unroll 8
    for (int k = 0; k < 128; ++k) acc = fmaf(hp[k], U_lstm[k * G_DIM + j], acc);
    biasrow[idx] = acc;
}

// ---- kernel 1: gates = X2d(8192x81) @ W_lstm(81x512) + biasrow (fp32 WMMA) ---
__global__ __launch_bounds__(128)
void gates_wmma_kernel(const float* __restrict__ X, const float* __restrict__ W,
                       const float* __restrict__ biasrow, float* __restrict__ gates) {
    const int lane = threadIdx.x & 31;
    const int wid  = threadIdx.x >> 5;
    const int n0 = (blockIdx.x * 4 + wid) * 16;   // N tile (0..31)
    const int m0 = blockIdx.y * 16;               // M tile (0..511)
    const int r  = lane & 15;
    const int kp = (lane >> 4) << 1;
    v8f acc = {};
    for (int k = 0; k < N_DIM; k += 4) {
        int k0 = k + kp, k1 = k0 + 1;
        v2f a, bf;
        a.x  = (k0 < N_DIM) ? X[(m0 + r) * N_DIM + k0] : 0.0f;
        a.y  = (k1 < N_DIM) ? X[(m0 + r) * N_DIM + k1] : 0.0f;
        bf.x = (k0 < N_DIM) ? W[k0 * G_DIM + n0 + r] : 0.0f;
        bf.y = (k1 < N_DIM) ? W[k1 * G_DIM + n0 + r] : 0.0f;
        acc = __builtin_amdgcn_wmma_f32_16x16x4_f32(false, a, false, bf, (short)0, acc, false, false);
    }
    const int col = n0 + r;
    const int mb  = m0 + ((lane >> 4) << 3);
    #pragma unroll
    for (int v = 0; v < 8; ++v) {
        int mm = mb + v;
        gates[(size_t)mm * G_DIM + col] = acc[v] + biasrow[(mm >> 6) * G_DIM + col];
    }
}

// ---- kernel 2: LSTM cell -> hc = [h | c]  (8192 x 256) ----------------------
__global__ __launch_bounds__(256)
void lstm_cell_kernel(const float* __restrict__ gates, const float* __restrict__ s0,
                      float* __restrict__ hc) {
    int idx = blockIdx.x * blockDim.x + threadIdx.x;   // 0..BT*128-1
    int bt = idx >> 7, mm = idx & 127;
    const float* g = gates + (size_t)bt * G_DIM;
    float i = sigm_(g[mm]);
    float f = sigm_(g[128 + mm]);
    float cg = tanhf(g[256 + mm]);
    float o = sigm_(g[384 + mm]);
    float c = f * s0[(bt >> 6) * 128 + mm] + i * cg;
    float h = o * tanhf(c);
    hc[(size_t)bt * 256 + mm]       = h;
    hc[(size_t)bt * 256 + 128 + mm] = c;
}

// ---- kernel 3: Aatt = hc(8192x256) @ We_w(256x64) + We_b (fp32 WMMA) --------
__global__ __launch_bounds__(128)
void attA_wmma_kernel(const float* __restrict__ hc, const float* __restrict__ Wew,
                      const float* __restrict__ Web, float* __restrict__ Aatt) {
    const int lane = threadIdx.x & 31;
    const int wid  = threadIdx.x >> 5;   // N tile 0..3
    const int m0 = blockIdx.x * 16;
    const int n0 = wid * 16;
    const int r  = lane & 15;
    const int kp = (lane >> 4) << 1;
    v8f acc = {};
    for (int k = 0; k < 256; k += 4) {
        int k0 = k + kp;
        v2f a, bf;
        a.x  = hc[(size_t)(m0 + r) * 256 + k0];
        a.y  = hc[(size_t)(m0 + r) * 256 + k0 + 1];
        bf.x = Wew[k0 * 64 + n0 + r];
        bf.y = Wew[(k0 + 1) * 64 + n0 + r];
        acc = __builtin_amdgcn_wmma_f32_16x16x4_f32(false, a, false, bf, (short)0, acc, false, false);
    }
    const int col = n0 + r;
    const int mb  = m0 + ((lane >> 4) << 3);
    float bias = Web[col];
    #pragma unroll
    for (int v = 0; v < 8; ++v)
        Aatt[(size_t)(mb + v) * 64 + col] = acc[v] + bias;
}

// ---- kernel 4: Ut[b][k][i] = (X[b]^T @ Ue_w)[i,k] + Ue_b[k]  (fp32 WMMA) ----
// Stored k-major so the score kernel's k-loop loads are coalesced over i.
__global__ __launch_bounds__(128)
void attU_wmma_kernel(const float* __restrict__ X, const float* __restrict__ Uew,
                      const float* __restrict__ Ueb, float* __restrict__ Ut) {
    const int lane = threadIdx.x & 31;
    const int wid  = threadIdx.x >> 5;   // N(k) tile 0..3
    const int b    = blockIdx.y;
    const int m0   = blockIdx.x * 16;    // i tile 0..5 (81 padded to 96)
    const int n0   = wid * 16;
    const int r    = lane & 15;
    const int kp   = (lane >> 4) << 1;
    const int gi   = m0 + r;
    const float* Xb = X + (size_t)b * ROWE;
    v8f acc = {};
    for (int t = 0; t < T_DIM; t += 4) {
        int t0 = t + kp;
        v2f a, bf;
        a.x  = (gi < N_DIM) ? Xb[t0 * N_DIM + gi] : 0.0f;
        a.y  = (gi < N_DIM) ? Xb[(t0 + 1) * N_DIM + gi] : 0.0f;
        bf.x = Uew[t0 * 64 + n0 + r];
        bf.y = Uew[(t0 + 1) * 64 + n0 + r];
        acc = __builtin_amdgcn_wmma_f32_16x16x4_f32(false, a, false, bf, (short)0, acc, false, false);
    }
    const int kcol = n0 + r;
    const int mb   = m0 + ((lane >> 4) << 3);
    float bias = Ueb[kcol];
    float* outp = Ut + (size_t)b * T_DIM * N_DIM;
    #pragma unroll
    for (int v = 0; v < 8; ++v) {
        int i = mb + v;
        if (i < N_DIM) outp[kcol * N_DIM + i] = acc[v] + bias;
    }
}

// ---- kernel 5: scores + softmax -> alpha (block per (b,t)) ------------------
__global__ __launch_bounds__(128)
void score_softmax_kernel(const float* __restrict__ Aatt, const float* __restrict__ Ut,
                          const float* __restrict__ vew, const float* __restrict__ veb,
                          float* __restrict__ alpha) {
    __shared__ float Arow[64];
    __shared__ float vsh[64];
    __shared__ float red[128];
    const int tid = threadIdx.x;
    const int t = blockIdx.x, b = blockIdx.y;
    const int bt = b * 64 + t;
    if (tid < 64) { Arow[tid] = Aatt[(size_t)bt * 64 + tid]; vsh[tid] = vew[tid]; }
    __syncthreads();
    float score = -3.0e38f;
    if (tid < N_DIM) {
        float acc = veb[0];
        const float* up = Ut + (size_t)b * T_DIM * N_DIM + tid;
        #pragma unroll 4
        for (int k = 0; k < 64; ++k)
            acc = fmaf(tanhf(Arow[k] + up[k * N_DIM]), vsh[k], acc);
        score = acc;
    }
    red[tid] = score; __syncthreads();
    for (int s = 64; s > 0; s >>= 1) {
        if (tid < s) red[tid] = fmaxf(red[tid], red[tid + s]);
        __syncthreads();
    }
    float mx = red[0]; __syncthreads();
    float ex = (tid < N_DIM) ? __expf(score - mx) : 0.0f;
    red[tid] = ex; __syncthreads();
    for (int s = 64; s > 0; s >>= 1) {
        if (tid < s) red[tid] += red[tid + s];
        __syncthreads();
    }
    float inv = 1.0f / red[0];
    if (tid < N_DIM) alpha[(size_t)bt * N_DIM + tid] = ex * inv;
}

// ---- kernel 6: out[t1,b,t2,i] = X[b,t2,i]*alpha[b,t1,i]  (streaming B128) ---
__global__ __launch_bounds__(256)
void broadcast_out_kernel(const float* __restrict__ X, const float* __restrict__ alpha,
                          float* __restrict__ out) {
    const int t1 = blockIdx.x, b = blockIdx.y;
    __shared__ float al[N_DIM];
    if (threadIdx.x < N_DIM)
        al[threadIdx.x] = alpha[((size_t)b * 64 + t1) * N_DIM + threadIdx.x];
    __syncthreads();
    const float4* Xv = reinterpret_cast<const float4*>(X + (size_t)b * ROWE);
    float4*       Ov = reinterpret_cast<float4*>(out + ((size_t)t1 * 128 + b) * ROWE);
    for (int j = threadIdx.x; j < ROWE / 4; j += 256) {
        float4 x = Xv[j];
        int i0 = (j * 4) % N_DIM;
        int i1 = i0 + 1; if (i1 >= N_DIM) i1 -= N_DIM;
        int i2 = i1 + 1; if (i2 >= N_DIM) i2 -= N_DIM;
        int i3 = i2 + 1; if (i3 >= N_DIM) i3 -= N_DIM;
        float4 r;
        r.x = x.x * al[i0]; r.y = x.y * al[i1];
        r.z = x.z * al[i2]; r.w = x.w * al[i3];
        Ov[j] = r;
    }
}

extern "C" void kernel_launch(void* const* d_in, const int* in_sizes, int n_in,
                              void* d_out, int out_size, void* d_ws, size_t ws_size,
                              hipStream_t stream) {
    (void)in_sizes; (void)n_in; (void)out_size; (void)ws_size;
    const float* X      = (const float*)d_in[0];
    const float* h0     = (const float*)d_in[1];
    const float* s0     = (const float*)d_in[2];
    const float* W_lstm = (const float*)d_in[3];
    const float* U_lstm = (const float*)d_in[4];
    const float* b_lstm = (const float*)d_in[5];
    const float* We_w   = (const float*)d_in[6];
    const float* We_b   = (const float*)d_in[7];
    const float* Ue_w   = (const float*)d_in[8];
    const float* Ue_b   = (const float*)d_in[9];
    const float* ve_w   = (const float*)d_in[10];
    const float* ve_b   = (const float*)d_in[11];
    float* out = (float*)d_out;

    float* ws      = (float*)d_ws;
    float* biasrow = ws;                       // 128*512      =    65,536
    float* gates   = biasrow + 65536;          // 8192*512     = 4,194,304
    float* hc      = gates + 4194304;          // 8192*256     = 2,097,152
    float* Aatt    = hc + 2097152;             // 8192*64      =   524,288
    float* Ut      = Aatt + 524288;            // 128*64*81    =   663,552
    float* alpha   = Ut + 663552;              // 128*64*81    =   663,552

    bias_gemm_kernel<<<256, 256, 0, stream>>>(h0, U_lstm, b_lstm, biasrow);
    gates_wmma_kernel<<<dim3(8, 512), 128, 0, stream>>>(X, W_lstm, biasrow, gates);
    lstm_cell_kernel<<<4096, 256, 0, stream>>>(gates, s0, hc);
    attA_wmma_kernel<<<512, 128, 0, stream>>>(hc, We_w, We_b, Aatt);
    attU_wmma_kernel<<<dim3(6, 128), 128, 0, stream>>>(X, Ue_w, Ue_b, Ut);
    score_softmax_kernel<<<dim3(64, 128), 128, 0, stream>>>(Aatt, Ut, ve_w, ve_b, alpha);
    broadcast_out_kernel<<<dim3(64, 128), 256, 0, stream>>>(X, alpha, out);
}